// GAT_74852690034751
// MI455X (gfx1250) — compile-verified
//
#include <hip/hip_runtime.h>

#define NN 100000            // nodes
#define NE 3200000           // edges (without self loops)
#define ET (NE + NN)         // total edges incl. self loops
#define INF_ 128             // in feats
#define HH 32                // heads*hidden (layer1 out width)
#define NHEADS 4
#define NSLOPE 0.2f

typedef __attribute__((ext_vector_type(2))) float v2f;
typedef __attribute__((ext_vector_type(8))) float v8f;

// ---- monotonic float <-> uint encoding for atomicMax-based segment max ----
__device__ __forceinline__ unsigned fenc(float f) {
  unsigned u = __float_as_uint(f);
  return (u >> 31) ? ~u : (u | 0x80000000u);
}
__device__ __forceinline__ float fdec(unsigned e) {
  return (e >> 31) ? __uint_as_float(e & 0x7FFFFFFFu) : __uint_as_float(~e);
}

__device__ __forceinline__ void edge_sd(const long long* __restrict__ ei, int e,
                                        int& s, int& d) {
  if (e < NE) { s = (int)ei[e]; d = (int)ei[(size_t)NE + e]; }
  else        { s = e - NE; d = s; }
}

// ---------------- Kernel 1: h1 = x @ W1  (fp32 WMMA 16x16x4) ----------------
// 8 waves per block; each wave owns a 16-row tile, 32 output columns.
__global__ __launch_bounds__(256) void k_gemm1(const float* __restrict__ x,
                                               const float* __restrict__ W1,
                                               float* __restrict__ h1) {
  __shared__ float sW[INF_ * HH];               // 16 KB
  const int tid  = threadIdx.x;
  const int wave = tid >> 5;
  const int lane = tid & 31;

  for (int i = tid; i < INF_ * HH; i += 256) sW[i] = W1[i];
  __syncthreads();

  const int tile = blockIdx.x * 8 + wave;
  const int row0 = tile * 16;
  const int m    = lane & 15;
  const int koff = (lane < 16) ? 0 : 2;         // K split across lane halves

  int arow = row0 + m;                          // clamp (stores are guarded)
  if (arow > NN - 1) arow = NN - 1;
  const float* xr = x + (size_t)arow * INF_;

  v8f c0 = {};
  v8f c1 = {};
  for (int ks = 0; ks < INF_; ks += 4) {
    v2f a;                                      // A 16x4: v0=K{0|2}, v1=K{1|3}
    a.x = xr[ks + koff];
    a.y = xr[ks + koff + 1];
    const float* wk = &sW[(ks + koff) * HH];    // B 4x16 mirrored layout
    v2f b0, b1;
    b0.x = wk[m];          b0.y = wk[HH + m];
    b1.x = wk[16 + m];     b1.y = wk[HH + 16 + m];
    c0 = __builtin_amdgcn_wmma_f32_16x16x4_f32(false, a, false, b0,
                                               (short)0, c0, false, false);
    c1 = __builtin_amdgcn_wmma_f32_16x16x4_f32(false, a, false, b1,
                                               (short)0, c1, false, false);
  }

  // C/D layout: VGPR i -> M = i (lanes 0-15) or 8+i (lanes 16-31); N = lane%16
  const int rbase = row0 + ((lane < 16) ? 0 : 8);
  for (int i = 0; i < 8; ++i) {
    int row = rbase + i;
    if (row < NN) {
      h1[(size_t)row * HH + m]      = c0[i];
      h1[(size_t)row * HH + 16 + m] = c1[i];
    }
  }
}

// ------------- Kernel 2: per-node attention coefficients (layer 1) ----------
__global__ void k_alpha1(const float* __restrict__ h1,
                         const float* __restrict__ a_src,   // [4,8]
                         const float* __restrict__ a_dst,   // [4,8]
                         float* __restrict__ as1, float* __restrict__ ad1) {
  int n = blockIdx.x * blockDim.x + threadIdx.x;
  if (n >= NN) return;
  const float* hp = h1 + (size_t)n * HH;
  for (int h = 0; h < NHEADS; ++h) {
    float s = 0.f, d = 0.f;
    for (int c = 0; c < 8; ++c) {
      float v = hp[h * 8 + c];
      s += v * a_src[h * 8 + c];
      d += v * a_dst[h * 8 + c];
    }
    as1[n * NHEADS + h] = s;
    ad1[n * NHEADS + h] = d;
  }
}

// ---------------- Kernel 3: segment max over dst (layer 1) ------------------
__global__ void k_emax1(const long long* __restrict__ ei,
                        const float* __restrict__ as1,
                        const float* __restrict__ ad1,
                        unsigned* __restrict__ m1) {
  int e = blockIdx.x * blockDim.x + threadIdx.x;
  if (e >= ET) return;
  int s, d; edge_sd(ei, e, s, d);
  for (int h = 0; h < NHEADS; ++h) {
    float v = as1[s * NHEADS + h] + ad1[d * NHEADS + h];
    v = v > 0.f ? v : NSLOPE * v;
    atomicMax(&m1[d * NHEADS + h], fenc(v));
  }
}

// ------- Kernel 4: exp-sum denominator + unnormalized aggregate (layer 1) ---
__global__ void k_esum1(const long long* __restrict__ ei,
                        const float* __restrict__ as1,
                        const float* __restrict__ ad1,
                        const unsigned* __restrict__ m1,
                        const float* __restrict__ h1,
                        float* __restrict__ den1,
                        float* __restrict__ agg1) {
  int e = blockIdx.x * blockDim.x + threadIdx.x;
  if (e >= ET) return;
  int s, d; edge_sd(ei, e, s, d);
  float ex[NHEADS];
  for (int h = 0; h < NHEADS; ++h) {
    float v = as1[s * NHEADS + h] + ad1[d * NHEADS + h];
    v = v > 0.f ? v : NSLOPE * v;
    float mx = fdec(m1[d * NHEADS + h]);
    ex[h] = __expf(v - mx);
    atomicAdd(&den1[d * NHEADS + h], ex[h]);
  }
  const float* hs = h1 + (size_t)s * HH;
  float* ag = agg1 + (size_t)d * HH;
  for (int j = 0; j < HH; ++j)
    atomicAdd(&ag[j], ex[j >> 3] * hs[j]);
}

// --- Kernel 5: normalize + bias + ELU, fused with layer-2 GEMV & alphas -----
__global__ void k_fin1(const float* __restrict__ agg1,
                       const float* __restrict__ den1,
                       const float* __restrict__ b1,
                       const float* __restrict__ W2,      // [32,1]
                       const float* __restrict__ a_s2,    // [1,1]
                       const float* __restrict__ a_d2,    // [1,1]
                       float* __restrict__ h2,
                       float* __restrict__ as2, float* __restrict__ ad2) {
  int n = blockIdx.x * blockDim.x + threadIdx.x;
  if (n >= NN) return;
  const float* ag = agg1 + (size_t)n * HH;
  float acc = 0.f;
  for (int j = 0; j < HH; ++j) {
    float v = ag[j] / den1[n * NHEADS + (j >> 3)] + b1[j];
    v = v > 0.f ? v : (__expf(v) - 1.0f);                 // ELU
    acc += v * W2[j];
  }
  h2[n]  = acc;
  as2[n] = acc * a_s2[0];
  ad2[n] = acc * a_d2[0];
}

// ---------------- Kernel 6/7: layer-2 edge passes (1 head) ------------------
__global__ void k_emax2(const long long* __restrict__ ei,
                        const float* __restrict__ as2,
                        const float* __restrict__ ad2,
                        unsigned* __restrict__ m2) {
  int e = blockIdx.x * blockDim.x + threadIdx.x;
  if (e >= ET) return;
  int s, d; edge_sd(ei, e, s, d);
  float v = as2[s] + ad2[d];
  v = v > 0.f ? v : NSLOPE * v;
  atomicMax(&m2[d], fenc(v));
}

__global__ void k_esum2(const long long* __restrict__ ei,
                        const float* __restrict__ as2,
                        const float* __restrict__ ad2,
                        const unsigned* __restrict__ m2,
                        const float* __restrict__ h2,
                        float* __restrict__ den2,
                        float* __restrict__ agg2) {
  int e = blockIdx.x * blockDim.x + threadIdx.x;
  if (e >= ET) return;
  int s, d; edge_sd(ei, e, s, d);
  float v = as2[s] + ad2[d];
  v = v > 0.f ? v : NSLOPE * v;
  float ex = __expf(v - fdec(m2[d]));
  atomicAdd(&den2[d], ex);
  atomicAdd(&agg2[d], ex * h2[s]);
}

// ---------------- Kernel 8: final output ------------------------------------
__global__ void k_fin2(const float* __restrict__ agg2,
                       const float* __restrict__ den2,
                       const float* __restrict__ b2,
                       float* __restrict__ out) {
  int n = blockIdx.x * blockDim.x + threadIdx.x;
  if (n >= NN) return;
  out[n] = agg2[n] / den2[n] + b2[0];
}

extern "C" void kernel_launch(void* const* d_in, const int* in_sizes, int n_in,
                              void* d_out, int out_size, void* d_ws, size_t ws_size,
                              hipStream_t stream) {
  (void)in_sizes; (void)n_in; (void)out_size; (void)ws_size;
  const float*     x    = (const float*)d_in[0];
  const long long* ei   = (const long long*)d_in[1];   // int64 [2, NE]
  const float*     W1   = (const float*)d_in[2];
  const float*     aS1  = (const float*)d_in[3];
  const float*     aD1  = (const float*)d_in[4];
  const float*     b1   = (const float*)d_in[5];
  const float*     W2   = (const float*)d_in[6];
  const float*     aS2  = (const float*)d_in[7];
  const float*     aD2  = (const float*)d_in[8];
  const float*     b2   = (const float*)d_in[9];
  float*           out  = (float*)d_out;

  char* ws = (char*)d_ws;
  size_t off = 0;
  auto alloc = [&](size_t bytes) -> void* {
    void* p = ws + off;
    off = (off + bytes + 255) & ~(size_t)255;
    return p;
  };
  float*    h1   = (float*)   alloc((size_t)NN * HH * 4);
  float*    as1  = (float*)   alloc((size_t)NN * NHEADS * 4);
  float*    ad1  = (float*)   alloc((size_t)NN * NHEADS * 4);
  float*    h2   = (float*)   alloc((size_t)NN * 4);
  float*    as2  = (float*)   alloc((size_t)NN * 4);
  float*    ad2  = (float*)   alloc((size_t)NN * 4);
  // ---- zero-initialized region (accumulators / maxima), contiguous ----
  size_t zoff = off;
  unsigned* m1   = (unsigned*)alloc((size_t)NN * NHEADS * 4);
  float*    den1 = (float*)   alloc((size_t)NN * NHEADS * 4);
  float*    agg1 = (float*)   alloc((size_t)NN * HH * 4);
  unsigned* m2   = (unsigned*)alloc((size_t)NN * 4);
  float*    den2 = (float*)   alloc((size_t)NN * 4);
  float*    agg2 = (float*)   alloc((size_t)NN * 4);
  size_t zbytes = off - zoff;
  hipMemsetAsync(ws + zoff, 0, zbytes, stream);   // graph-capture safe

  const int TPB = 256;
  const int tiles  = (NN + 15) / 16;              // 6250
  const int gblks  = (tiles + 7) / 8;             // 782
  const int nblks  = (NN + TPB - 1) / TPB;
  const int eblks  = (ET + TPB - 1) / TPB;

  k_gemm1 <<<gblks, TPB, 0, stream>>>(x, W1, h1);
  k_alpha1<<<nblks, TPB, 0, stream>>>(h1, aS1, aD1, as1, ad1);
  k_emax1 <<<eblks, TPB, 0, stream>>>(ei, as1, ad1, m1);
  k_esum1 <<<eblks, TPB, 0, stream>>>(ei, as1, ad1, m1, h1, den1, agg1);
  k_fin1  <<<nblks, TPB, 0, stream>>>(agg1, den1, b1, W2, aS2, aD2, h2, as2, ad2);
  k_emax2 <<<eblks, TPB, 0, stream>>>(ei, as2, ad2, m2);
  k_esum2 <<<eblks, TPB, 0, stream>>>(ei, as2, ad2, m2, h2, den2, agg2);
  k_fin2  <<<nblks, TPB, 0, stream>>>(agg2, den2, b2, out);
}